// Query_Guided_Attention_Layer_72370198938045
// MI455X (gfx1250) — compile-verified
//
#include <hip/hip_runtime.h>
#include <hip/hip_bf16.h>
#include <math.h>
#include <stdint.h>

// ---------------------------------------------------------------------------
// Query-guided attention layer, MI455X (gfx1250), f32 WMMA 16x16x4 pipeline
// with async global->LDS staging of activations (ASYNCcnt path).
// N=64, C=512, CI=256, H=12, W=4 (HW=48).
// Key simplification: segment_sum over gs[p]=j collapses the N^2 scatter to
//   x2[j,c,h,w] = x[j,c,h,w] * M[j,h,w],  M[j,s] = sum_i max_k f[i,j,s,k].
// ---------------------------------------------------------------------------

#define NN     64
#define C_     512
#define CI_    256
#define HW_    48
#define PPAD   16       // phi_p padded from 12 -> 16 cols (zeros)
#define CHW    (C_ * HW_)
#define KCHUNK 256      // K rows staged per LDS phase (256*48*4B = 48 KB)

typedef __attribute__((ext_vector_type(2))) float v2f;
typedef __attribute__((ext_vector_type(8))) float v8f;

__device__ __forceinline__ v8f wmma_f32(v2f a, v2f b, v8f c) {
  // V_WMMA_F32_16X16X4_F32 : D = A(16x4) x B(4x16) + C(16x16), all f32.
  return __builtin_amdgcn_wmma_f32_16x16x4_f32(
      false, a, false, b, (short)0, c, false, false);
}

// ---------------------------------------------------------------------------
// Async-stage 48 KB (KCHUNK x 48 floats) from global into LDS.
// Each of the 256 threads copies 12 x 16B slices; IOFFSET applies to both the
// LDS dest and the global source (ISA 15.18),
// so one (laddr, gaddr) pair covers the whole chunk via offset immediates.
// Tracked by ASYNCcnt; wave waits its own transfers, barrier publishes them.
// ---------------------------------------------------------------------------
__device__ __forceinline__ void stage_chunk_async(unsigned laddr,
                                                  const float* gptr, int t) {
  unsigned long long ga =
      (unsigned long long)(uintptr_t)gptr + (unsigned long long)t * 16ull;
  asm volatile(
      "global_load_async_to_lds_b128 %0, %1, off\n\t"
      "global_load_async_to_lds_b128 %0, %1, off offset:4096\n\t"
      "global_load_async_to_lds_b128 %0, %1, off offset:8192\n\t"
      "global_load_async_to_lds_b128 %0, %1, off offset:12288\n\t"
      "global_load_async_to_lds_b128 %0, %1, off offset:16384\n\t"
      "global_load_async_to_lds_b128 %0, %1, off offset:20480\n\t"
      "global_load_async_to_lds_b128 %0, %1, off offset:24576\n\t"
      "global_load_async_to_lds_b128 %0, %1, off offset:28672\n\t"
      "global_load_async_to_lds_b128 %0, %1, off offset:32768\n\t"
      "global_load_async_to_lds_b128 %0, %1, off offset:36864\n\t"
      "global_load_async_to_lds_b128 %0, %1, off offset:40960\n\t"
      "global_load_async_to_lds_b128 %0, %1, off offset:45056\n\t"
      "s_wait_asynccnt 0x0"
      :: "v"(laddr), "v"(ga)
      : "memory");
}

// ---------------------------------------------------------------------------
// K0: zero the attention accumulator
// ---------------------------------------------------------------------------
__global__ void zero_kernel(float* __restrict__ p, int n) {
  int i = blockIdx.x * blockDim.x + threadIdx.x;
  if (i < n) p[i] = 0.0f;
}

// ---------------------------------------------------------------------------
// K1: theta/phi projections.  8 waves/WG; each wave computes a 16(M) x 48(S)
// strip of  out[n] = W(256x512) @ x[n](512x48) + bias.  x[n] is staged into
// LDS in two 48KB K-chunks shared by all 8 waves (async global->LDS).
// theta stored transposed [N,48,256] (contiguous A operand for K3);
// phi stored [N,256,48] (row-major for pooling).
// ---------------------------------------------------------------------------
__global__ __launch_bounds__(256) void proj_kernel(
    const float* __restrict__ x,
    const float* __restrict__ theta_w, const float* __restrict__ theta_b,
    const float* __restrict__ phi_w,   const float* __restrict__ phi_b,
    float* __restrict__ theta_t,  // [N, 48, 256]
    float* __restrict__ phi)      // [N, 256, 48]
{
  __shared__ float smem[KCHUNK * HW_];   // 48 KB
  const int n     = blockIdx.x;
  const int mg    = blockIdx.y;          // 0..1 (group of 8 M-tiles)
  const int which = blockIdx.z;          // 0 = theta, 1 = phi
  const int t     = threadIdx.x;
  const int wave  = t >> 5;
  const int lane  = t & 31;
  const int lm = lane & 15, hi = lane >> 4;
  const int mt = mg * 8 + wave;          // 0..15

  const float* W  = which ? phi_w : theta_w;
  const float* Bv = which ? phi_b : theta_b;

  const float* wrow = W + (size_t)(mt * 16 + lm) * C_ + 2 * hi;
  const float* gsrc = x + (size_t)n * CHW;
  const unsigned lbase = (unsigned)(uintptr_t)(&smem[0]);

  v8f acc0 = {}, acc1 = {}, acc2 = {};
  for (int kc = 0; kc < C_; kc += KCHUNK) {
    stage_chunk_async(lbase + (unsigned)t * 16u, gsrc + (size_t)kc * HW_, t);
    __syncthreads();
    const float* bbase = smem + 2 * hi * HW_ + lm;
    for (int k0 = 0; k0 < KCHUNK; k0 += 4) {
      v2f a = *(const v2f*)(wrow + kc + k0);
      const float* xp = bbase + (size_t)k0 * HW_;
      v2f b0, b1, b2;
      b0.x = xp[0];  b0.y = xp[HW_];
      b1.x = xp[16]; b1.y = xp[HW_ + 16];
      b2.x = xp[32]; b2.y = xp[HW_ + 32];
      acc0 = wmma_f32(a, b0, acc0);
      acc1 = wmma_f32(a, b1, acc1);
      acc2 = wmma_f32(a, b2, acc2);
    }
    __syncthreads();   // all waves done reading before next stage overwrites
  }

  v8f accs[3] = {acc0, acc1, acc2};
  if (which == 0) {
    // theta_t[n, s, m] : lane holds col s, rows m = mt*16 + 8*hi + v
    for (int st = 0; st < 3; ++st) {
      int s = st * 16 + lm;
      float* dst = theta_t + ((size_t)n * HW_ + s) * CI_ + mt * 16 + 8 * hi;
#pragma unroll
      for (int v = 0; v < 8; ++v)
        dst[v] = accs[st][v] + Bv[mt * 16 + 8 * hi + v];
    }
  } else {
    for (int st = 0; st < 3; ++st) {
      int s = st * 16 + lm;
#pragma unroll
      for (int v = 0; v < 8; ++v) {
        int m = mt * 16 + 8 * hi + v;
        phi[((size_t)n * CI_ + m) * HW_ + s] = accs[st][v] + Bv[m];
      }
    }
  }
}

// ---------------------------------------------------------------------------
// K2: MaxPool2d(2,2): phi [N,256,12,4] -> phi_p [N,256,16] (cols 12..15 = 0)
// ---------------------------------------------------------------------------
__global__ void pool_kernel(const float* __restrict__ phi,
                            float* __restrict__ phi_p) {
  int idx = blockIdx.x * blockDim.x + threadIdx.x;
  if (idx >= NN * CI_ * PPAD) return;
  int p  = idx & 15;
  int nc = idx >> 4;             // n*CI_ + c
  float r = 0.0f;
  if (p < 12) {
    int ph = p >> 1, pw = p & 1;
    const float* base = phi + (size_t)nc * HW_ + (2 * ph) * 4 + 2 * pw;
    r = fmaxf(fmaxf(base[0], base[1]), fmaxf(base[4], base[5]));
  }
  phi_p[idx] = r;
}

// ---------------------------------------------------------------------------
// K3: pairwise scores + max + reduce-over-i.
// Wave handles (j, s-tile, chunk of 8 i's):
//   G = theta_t[j, s0:s0+16, :](16x256) @ phi_p[i](256x16)   (WMMA, K=256)
//   m = max over cols 0..11 of G/12  -> accumulate over i into Msum[j,s].
// phi_p[i] is 16KB and L0/L2-resident across the i-loop; no staging needed.
// ---------------------------------------------------------------------------
__global__ __launch_bounds__(32) void attn_kernel(
    const float* __restrict__ theta_t,  // [N,48,256]
    const float* __restrict__ phi_p,    // [N,256,16]
    float* __restrict__ Msum)           // [N,48]
{
  const int j    = blockIdx.x;
  const int st   = blockIdx.y;   // 0..2
  const int ic   = blockIdx.z;   // 0..7 (i chunk)
  const int lane = threadIdx.x;
  const int lm = lane & 15, hi = lane >> 4;

  const float* arow =
      theta_t + ((size_t)j * HW_ + st * 16 + lm) * CI_ + 2 * hi;

  float msum[8];
#pragma unroll
  for (int v = 0; v < 8; ++v) msum[v] = 0.0f;

  for (int ii = 0; ii < 8; ++ii) {
    int i = ic * 8 + ii;
    const float* bb = phi_p + ((size_t)i * CI_ + 2 * hi) * PPAD + lm;
    v8f acc = {};
    for (int k0 = 0; k0 < CI_; k0 += 4) {
      v2f a = *(const v2f*)(arow + k0);
      const float* bp = bb + (size_t)k0 * PPAD;
      v2f b; b.x = bp[0]; b.y = bp[PPAD];
      acc = wmma_f32(a, b, acc);
    }
    // max over valid pooled positions (cols 0..11), per row (v, v+8)
#pragma unroll
    for (int v = 0; v < 8; ++v) {
      float val = (lm < 12) ? acc[v] * (1.0f / 12.0f) : -1e30f;
      val = fmaxf(val, __shfl_xor(val, 1, 32));
      val = fmaxf(val, __shfl_xor(val, 2, 32));
      val = fmaxf(val, __shfl_xor(val, 4, 32));
      val = fmaxf(val, __shfl_xor(val, 8, 32));
      msum[v] += val;   // all 16 lanes of the half hold the row max
    }
  }
#pragma unroll
  for (int v = 0; v < 8; ++v) {
    if (lm == v)
      atomicAdd(&Msum[j * HW_ + st * 16 + v + 8 * hi], msum[v]);
  }
}

// ---------------------------------------------------------------------------
// K4/K6: per-sample LayerNorm over (C,H,W); optionally applies the attention
// scaling val = x * (1 + Msum[j,s]) first (res1 fused).
// ---------------------------------------------------------------------------
__global__ __launch_bounds__(256) void ln_kernel(
    const float* __restrict__ vin,   // [N,512,48]
    const float* __restrict__ msum,  // [N,48] or nullptr
    const float* __restrict__ g, const float* __restrict__ b,
    float* __restrict__ out)         // [N,512,48]
{
  __shared__ float s_sum[256], s_sq[256];
  __shared__ float s_mu, s_rstd;
  const int n = blockIdx.x;
  const int t = threadIdx.x;
  const float* vp = vin + (size_t)n * CHW;

  float sum = 0.f, sq = 0.f;
  for (int idx = t; idx < CHW; idx += 256) {
    float val = vp[idx];
    if (msum) val *= (1.0f + msum[n * HW_ + (idx % HW_)]);
    sum += val; sq += val * val;
  }
  s_sum[t] = sum; s_sq[t] = sq;
  __syncthreads();
  for (int off = 128; off > 0; off >>= 1) {
    if (t < off) { s_sum[t] += s_sum[t + off]; s_sq[t] += s_sq[t + off]; }
    __syncthreads();
  }
  if (t == 0) {
    float mu  = s_sum[0] / (float)CHW;
    float var = s_sq[0] / (float)CHW - mu * mu;
    s_mu = mu; s_rstd = rsqrtf(var + 1e-5f);
  }
  __syncthreads();
  float mu = s_mu, rstd = s_rstd;
  for (int idx = t; idx < CHW; idx += 256) {
    float val = vp[idx];
    if (msum) val *= (1.0f + msum[n * HW_ + (idx % HW_)]);
    out[(size_t)n * CHW + idx] = (val - mu) * rstd * g[idx] + b[idx];
  }
}

// ---------------------------------------------------------------------------
// K5: 512x512 1x1-conv GEMM per sample, optional relu / residual add.
//   Out[n] = act( W(512x512) @ In[n](512x48) ) [+ Res[n]]
// 8 waves/WG sharing the async-staged activation chunk in LDS.
// ---------------------------------------------------------------------------
__global__ __launch_bounds__(256) void mlp_gemm(
    const float* __restrict__ W,    // [512,512]
    const float* __restrict__ In,   // [N,512,48]
    const float* __restrict__ Res,  // [N,512,48] or nullptr
    float* __restrict__ Out,        // [N,512,48]
    int do_relu)
{
  __shared__ float smem[KCHUNK * HW_];   // 48 KB
  const int n    = blockIdx.x;
  const int mg   = blockIdx.y;           // 0..3 (group of 8 M-tiles)
  const int t    = threadIdx.x;
  const int wave = t >> 5;
  const int lane = t & 31;
  const int lm = lane & 15, hi = lane >> 4;
  const int mt = mg * 8 + wave;          // 0..31

  const float* wrow = W + (size_t)(mt * 16 + lm) * C_ + 2 * hi;
  const float* gsrc = In + (size_t)n * CHW;
  const unsigned lbase = (unsigned)(uintptr_t)(&smem[0]);

  v8f acc0 = {}, acc1 = {}, acc2 = {};
  for (int kc = 0; kc < C_; kc += KCHUNK) {
    stage_chunk_async(lbase + (unsigned)t * 16u, gsrc + (size_t)kc * HW_, t);
    __syncthreads();
    const float* bbase = smem + 2 * hi * HW_ + lm;
    for (int k0 = 0; k0 < KCHUNK; k0 += 4) {
      v2f a = *(const v2f*)(wrow + kc + k0);
      const float* xp = bbase + (size_t)k0 * HW_;
      v2f b0, b1, b2;
      b0.x = xp[0];  b0.y = xp[HW_];
      b1.x = xp[16]; b1.y = xp[HW_ + 16];
      b2.x = xp[32]; b2.y = xp[HW_ + 32];
      acc0 = wmma_f32(a, b0, acc0);
      acc1 = wmma_f32(a, b1, acc1);
      acc2 = wmma_f32(a, b2, acc2);
    }
    __syncthreads();
  }

  v8f accs[3] = {acc0, acc1, acc2};
  for (int st = 0; st < 3; ++st) {
    int s = st * 16 + lm;
#pragma unroll
    for (int v = 0; v < 8; ++v) {
      int m = mt * 16 + 8 * hi + v;
      size_t o = ((size_t)n * C_ + m) * HW_ + s;
      float r = accs[st][v];
      if (do_relu) r = fmaxf(r, 0.0f);
      if (Res) r += Res[o];
      Out[o] = r;
    }
  }
}

// ---------------------------------------------------------------------------
// Host-side launch
// ---------------------------------------------------------------------------
extern "C" void kernel_launch(void* const* d_in, const int* in_sizes, int n_in,
                              void* d_out, int out_size, void* d_ws, size_t ws_size,
                              hipStream_t stream) {
  const float* x       = (const float*)d_in[0];
  const float* theta_w = (const float*)d_in[1];
  const float* theta_b = (const float*)d_in[2];
  const float* phi_w   = (const float*)d_in[3];
  const float* phi_b   = (const float*)d_in[4];
  const float* conv1_w = (const float*)d_in[5];
  const float* conv2_w = (const float*)d_in[6];
  const float* ln1_g   = (const float*)d_in[7];
  const float* ln1_b   = (const float*)d_in[8];
  const float* ln2_g   = (const float*)d_in[9];
  const float* ln2_b   = (const float*)d_in[10];
  float* out = (float*)d_out;

  float* ws      = (float*)d_ws;
  float* theta_t = ws;                               // 64*48*256
  float* phi     = theta_t + (size_t)NN * HW_ * CI_; // 64*256*48
  float* phi_p   = phi + (size_t)NN * CI_ * HW_;     // 64*256*16
  float* Msum    = phi_p + (size_t)NN * CI_ * PPAD;  // 64*48
  float* x1      = Msum + (size_t)NN * HW_;          // 64*512*48
  float* mid     = x1 + (size_t)NN * CHW;            // 64*512*48
  float* y2      = mid + (size_t)NN * CHW;           // 64*512*48

  zero_kernel<<<(NN * HW_ + 255) / 256, 256, 0, stream>>>(Msum, NN * HW_);
  proj_kernel<<<dim3(NN, 2, 2), 256, 0, stream>>>(
      x, theta_w, theta_b, phi_w, phi_b, theta_t, phi);
  pool_kernel<<<(NN * CI_ * PPAD + 255) / 256, 256, 0, stream>>>(phi, phi_p);
  attn_kernel<<<dim3(NN, 3, 8), 32, 0, stream>>>(theta_t, phi_p, Msum);
  ln_kernel<<<NN, 256, 0, stream>>>(x, Msum, ln1_g, ln1_b, x1);
  mlp_gemm<<<dim3(NN, 4), 256, 0, stream>>>(conv1_w, x1, nullptr, mid, 1);
  mlp_gemm<<<dim3(NN, 4), 256, 0, stream>>>(conv2_w, mid, x1, y2, 0);
  ln_kernel<<<NN, 256, 0, stream>>>(y2, nullptr, ln2_g, ln2_b, out);
}